// AttentionMaskGenerator_35579509080124
// MI455X (gfx1250) — compile-verified
//
#include <hip/hip_runtime.h>
#include <math.h>

// Problem constants from the reference setup.
#define S_LEN     2048
#define D_DIM     128
#define NTYPES    3
#define NUM_HEADS 8
#define EPS_G     1e-10f

typedef float v2f __attribute__((ext_vector_type(2)));
typedef float v4f __attribute__((ext_vector_type(4)));
typedef float v8f __attribute__((ext_vector_type(8)));

// ---------------------------------------------------------------------------
// Kernel A: per-batch  logits (WMMA f32 16x16x4)  ->  gumbel argmax (tt)
//           ->  reverse min-scan (next_global).  One workgroup per batch.
// ---------------------------------------------------------------------------
__global__ __launch_bounds__(1024)
void tt_scan_kernel(const float* __restrict__ input,     // (B,S,D)
                    const float* __restrict__ gumbel_u,  // (B,S,3)
                    const float* __restrict__ W,         // (3,D)
                    const float* __restrict__ bias,      // (3,)
                    int* __restrict__ tt_out,            // (B,S)
                    int* __restrict__ ng_out)            // (B,S)
{
    __shared__ float s_logits[S_LEN * NTYPES];  // 24 KB
    __shared__ int   s_gidx[S_LEN];             //  8 KB
    __shared__ int   s_tt[S_LEN];               //  8 KB

    const int b    = blockIdx.x;
    const int tid  = threadIdx.x;
    const int wave = tid >> 5;
    const int lane = tid & 31;
    const int half = lane >> 4;   // 0: K={0,1}, 1: K={2,3}  (32-bit A/B layout)
    const int lm   = lane & 15;   // A: row-in-tile, B: column N

    const float* inB = input + (size_t)b * S_LEN * D_DIM;

    // ---- Phase 1: logits[s, t] = input[s,:] . W[t,:]  via V_WMMA_F32_16X16X4_F32
    const int  wn     = (lm < NTYPES) ? lm : (NTYPES - 1);
    const bool bvalid = (lm < NTYPES);
    const float* brow = W + (size_t)wn * D_DIM;

    for (int tile = wave; tile < S_LEN / 16; tile += 32) {
        const int row = tile * 16 + lm;
        const float* arow = inB + (size_t)row * D_DIM;
        v8f c = {};
        for (int kc = 0; kc < D_DIM / 4; ++kc) {
            const int k = kc * 4 + 2 * half;
            v2f a;  a.x = arow[k];  a.y = arow[k + 1];
            v2f bv;
            const float wb0 = brow[k], wb1 = brow[k + 1];
            bv.x = bvalid ? wb0 : 0.0f;
            bv.y = bvalid ? wb1 : 0.0f;
            c = __builtin_amdgcn_wmma_f32_16x16x4_f32(
                    /*neg_a=*/false, a, /*neg_b=*/false, bv,
                    /*c_mod=*/(short)0, c, /*reuse_a=*/false, /*reuse_b=*/false);
        }
        // D layout: lane<16 holds N=lane, rows M=vgpr 0..7; lane>=16 rows 8..15.
        if (lm < NTYPES) {
            #pragma unroll
            for (int m = 0; m < 8; ++m) {
                const int rl = tile * 16 + m + 8 * half;
                s_logits[rl * NTYPES + lm] = c[m];
            }
        }
    }
    __syncthreads();

    // ---- Phase 2: tt = argmax_t(logits + bias + gumbel)   (tau == 1)
    const float bb0 = bias[0], bb1 = bias[1], bb2 = bias[2];
    for (int s = tid; s < S_LEN; s += 1024) {
        const float* u = gumbel_u + ((size_t)b * S_LEN + s) * NTYPES;
        const float g0 = -logf(-logf(u[0] + EPS_G) + EPS_G);
        const float g1 = -logf(-logf(u[1] + EPS_G) + EPS_G);
        const float g2 = -logf(-logf(u[2] + EPS_G) + EPS_G);
        const float v0 = s_logits[s * 3 + 0] + bb0 + g0;
        const float v1 = s_logits[s * 3 + 1] + bb1 + g1;
        const float v2 = s_logits[s * 3 + 2] + bb2 + g2;
        int t = 0; float best = v0;                 // first-max tie-break
        if (v1 > best) { best = v1; t = 1; }
        if (v2 > best) { t = 2; }
        s_tt[s]   = t;
        s_gidx[s] = (t == 0) ? s : S_LEN;
    }
    __syncthreads();

    // ---- Phase 3: reverse inclusive min-scan (Hillis-Steele, 11 steps)
    for (int off = 1; off < S_LEN; off <<= 1) {
        const int s1 = tid, s2 = tid + 1024;
        const int a1  = s_gidx[s1];
        const int a1n = (s1 + off < S_LEN) ? s_gidx[s1 + off] : S_LEN;
        const int a2  = s_gidx[s2];
        const int a2n = (s2 + off < S_LEN) ? s_gidx[s2 + off] : S_LEN;
        __syncthreads();
        s_gidx[s1] = min(a1, a1n);
        s_gidx[s2] = min(a2, a2n);
        __syncthreads();
    }

    // next_global[s] = suffix_min[s+1] (S at the end)
    for (int s = tid; s < S_LEN; s += 1024) {
        tt_out[b * S_LEN + s] = s_tt[s];
        ng_out[b * S_LEN + s] = (s + 1 < S_LEN) ? s_gidx[s + 1] : S_LEN;
    }
}

// ---------------------------------------------------------------------------
// Kernel B: streaming mask writer.  One block per (b, i) row; each float4 is
// computed once and stored to all 8 identical head slices with NT b128 stores.
// ---------------------------------------------------------------------------
__global__ __launch_bounds__(256)
void mask_write_kernel(const int* __restrict__ tt_in,
                       const int* __restrict__ ng_in,
                       float* __restrict__ out)
{
    const int bi = blockIdx.x;              // b * S + i
    const int b  = bi >> 11;                // / 2048
    const int i  = bi & (S_LEN - 1);
    const int tt = tt_in[bi];
    const int ng = ng_in[bi];

    // tt==0: j<=i ; tt==2: j<=min(i,ng) ; tt==1: j==i (window & causal collapse)
    int limit = (tt == 2) ? min(i, ng) : i;
    const bool diag_only = (tt == 1);

    float* base = out + ((size_t)b * NUM_HEADS) * (size_t)S_LEN * S_LEN
                      + (size_t)i * S_LEN;
    const size_t hs = (size_t)S_LEN * S_LEN;

    for (int j0 = threadIdx.x * 4; j0 < S_LEN; j0 += 256 * 4) {
        v4f v;
        if (diag_only) {
            v.x = (j0 + 0 == i) ? 1.0f : 0.0f;
            v.y = (j0 + 1 == i) ? 1.0f : 0.0f;
            v.z = (j0 + 2 == i) ? 1.0f : 0.0f;
            v.w = (j0 + 3 == i) ? 1.0f : 0.0f;
        } else {
            v.x = (j0 + 0 <= limit) ? 1.0f : 0.0f;
            v.y = (j0 + 1 <= limit) ? 1.0f : 0.0f;
            v.z = (j0 + 2 <= limit) ? 1.0f : 0.0f;
            v.w = (j0 + 3 <= limit) ? 1.0f : 0.0f;
        }
        #pragma unroll
        for (int h = 0; h < NUM_HEADS; ++h) {
            __builtin_nontemporal_store(v, (v4f*)(base + h * hs + j0));
        }
    }
}

// ---------------------------------------------------------------------------
extern "C" void kernel_launch(void* const* d_in, const int* in_sizes, int n_in,
                              void* d_out, int out_size, void* d_ws, size_t ws_size,
                              hipStream_t stream)
{
    const float* input  = (const float*)d_in[0];   // (B,S,D) f32
    // d_in[1] = token_types (unused by the reference)
    const float* gumbel = (const float*)d_in[2];   // (B,S,3) f32
    const float* W      = (const float*)d_in[3];   // (3,D)   f32
    const float* bias   = (const float*)d_in[4];   // (3,)    f32
    float* out = (float*)d_out;                    // (B,8,S,S) f32

    const int B = in_sizes[1] / S_LEN;             // token_types is (B,S)

    int* tt_ws = (int*)d_ws;
    int* ng_ws = tt_ws + B * S_LEN;

    tt_scan_kernel<<<B, 1024, 0, stream>>>(input, gumbel, W, bias, tt_ws, ng_ws);
    mask_write_kernel<<<B * S_LEN, 256, 0, stream>>>(tt_ws, ng_ws, out);
}